// GCN_Linear_19275813225180
// MI455X (gfx1250) — compile-verified
//
#include <hip/hip_runtime.h>
#include <hip/hip_bf16.h>
#include <math.h>

// ---------------------------------------------------------------------------
// GCN (2x GCNConv + linear + log_softmax) for MI455X (gfx1250, wave32, WMMA)
//   N=170000 nodes, E=1200000 edges, F_IN=128, H=64, C=40
// Memory/scatter-bound op: keep fp32 end-to-end; V_WMMA_F32_16X16X4_F32 GEMMs.
// ---------------------------------------------------------------------------

typedef float v2f __attribute__((ext_vector_type(2)));
typedef float v8f __attribute__((ext_vector_type(8)));

#define F_IN 128
#define HDIM 64
#define CDIM 40

// ---------------------------------------------------------------------------
// deg[d] += ew[e]  (self-loop +1 applied in dinv kernel)
// ---------------------------------------------------------------------------
__global__ void deg_kernel(const int* __restrict__ dst, const float* __restrict__ ew,
                           float* __restrict__ deg, int E) {
  int e = blockIdx.x * blockDim.x + threadIdx.x;
  if (e < E) unsafeAtomicAdd(&deg[dst[e]], ew[e]);
}

__global__ void dinv_kernel(const float* __restrict__ deg, float* __restrict__ dinv, int N) {
  int n = blockIdx.x * blockDim.x + threadIdx.x;
  if (n < N) {
    float d = deg[n] + 1.0f;              // add self-loop weight
    dinv[n] = (d > 0.0f) ? rsqrtf(d) : 0.0f;
  }
}

// ---------------------------------------------------------------------------
// Y[M x NO] = X[M x KD] @ W[KD x NO]      (row-major, no bias)
// One wave computes a 16-row x (NTILES*16)-col tile with V_WMMA_F32_16X16X4_F32.
//
// W is staged TRANSPOSED + zero-padded in LDS: WT[n][k], n in [0, NTILES*16),
// row stride KD+2 (bank de-skew). This makes all inner-loop loads
// unconditional: A = one global_load_b64/lane, B = one ds_load_b64/lane.
//
// fp32 WMMA operand layout (ISA 7.12.2, 32-bit 16x4 A):
//   A (16x4): lane%16 = M; lanes<16 hold K={kb,kb+1}, lanes>=16 hold K={kb+2,kb+3}
//   B (4x16): lane%16 = N; same K split (row-striped analog of C/D)
//   C/D (16x16): VGPR r -> M = r + 8*(lane/16), N = lane%16
// Tail safety: A row index is clamped (garbage only lands in D rows >= M,
// which are never stored); B cols >= NO are zero-padded in LDS.
// ---------------------------------------------------------------------------
template <int NTILES, int KD>
__global__ void __launch_bounds__(256)
gemm_wmma_f32(const float* __restrict__ X, const float* __restrict__ W,
              float* __restrict__ Y, int M, int NO) {
  constexpr int NCOLS  = NTILES * 16;
  constexpr int STRIDE = KD + 2;
  __shared__ float WT[NCOLS * STRIDE];

  const int tid = threadIdx.x;
  // Stage W^T (zero-padded cols) into LDS.
  for (int i = tid; i < NCOLS * KD; i += 256) {
    const int n = i / KD;
    const int k = i - n * KD;
    WT[n * STRIDE + k] = (n < NO) ? W[k * NO + n] : 0.0f;
  }
  __syncthreads();

  const int wave = tid >> 5;
  const int lane = tid & 31;
  const int half = lane >> 4;   // 0: lanes 0-15, 1: lanes 16-31
  const int l16  = lane & 15;

  const int tile = blockIdx.x * 8 + wave;        // 16-row tile index
  const int mrow = tile * 16 + l16;              // logical A row for this lane
  const int mclk = (mrow < M) ? mrow : 0;        // clamped (no divergent loads)
  const float* __restrict__ xrow = X + (size_t)mclk * KD;

  v8f acc0 = {}, acc1 = {}, acc2 = {}, acc3 = {};

#pragma unroll 4
  for (int kb = 0; kb < KD; kb += 4) {
    const int ka = kb + 2 * half;                // even -> 8B aligned
    const v2f a = *reinterpret_cast<const v2f*>(xrow + ka);

    {
      const v2f b = *reinterpret_cast<const v2f*>(&WT[(0 * 16 + l16) * STRIDE + ka]);
      acc0 = __builtin_amdgcn_wmma_f32_16x16x4_f32(false, a, false, b, (short)0, acc0, false, false);
    }
    if constexpr (NTILES > 1) {
      const v2f b = *reinterpret_cast<const v2f*>(&WT[(1 * 16 + l16) * STRIDE + ka]);
      acc1 = __builtin_amdgcn_wmma_f32_16x16x4_f32(false, a, false, b, (short)0, acc1, false, false);
    }
    if constexpr (NTILES > 2) {
      const v2f b = *reinterpret_cast<const v2f*>(&WT[(2 * 16 + l16) * STRIDE + ka]);
      acc2 = __builtin_amdgcn_wmma_f32_16x16x4_f32(false, a, false, b, (short)0, acc2, false, false);
    }
    if constexpr (NTILES > 3) {
      const v2f b = *reinterpret_cast<const v2f*>(&WT[(3 * 16 + l16) * STRIDE + ka]);
      acc3 = __builtin_amdgcn_wmma_f32_16x16x4_f32(false, a, false, b, (short)0, acc3, false, false);
    }
  }

  // Store D. Row r of accumulator VGPR maps to M = tile*16 + r + 8*half.
  const int rbase = tile * 16 + 8 * half;
#pragma unroll
  for (int r = 0; r < 8; ++r) {
    const int row = rbase + r;
    if (row < M) {
      float* __restrict__ yrow = Y + (size_t)row * NO;
      {
        const int n = 0 * 16 + l16;
        if (n < NO) yrow[n] = acc0[r];
      }
      if constexpr (NTILES > 1) {
        const int n = 1 * 16 + l16;
        if (n < NO) yrow[n] = acc1[r];
      }
      if constexpr (NTILES > 2) {
        const int n = 2 * 16 + l16;
        if (n < NO) yrow[n] = acc2[r];
      }
      if constexpr (NTILES > 3) {
        const int n = 3 * 16 + l16;
        if (n < NO) yrow[n] = acc3[r];
      }
    }
  }
}

// ---------------------------------------------------------------------------
// agg[dst] += dinv[src]*ew*dinv[dst] * h[src]   (32 lanes per edge, float2/lane)
// ---------------------------------------------------------------------------
__global__ void scatter_kernel(const int* __restrict__ src, const int* __restrict__ dst,
                               const float* __restrict__ ew, const float* __restrict__ dinv,
                               const float* __restrict__ h, float* __restrict__ agg, int E) {
  int t = blockIdx.x * blockDim.x + threadIdx.x;
  int e = t >> 5;
  if (e >= E) return;
  int lane = t & 31;
  int s = src[e];
  int d = dst[e];
  float norm = dinv[s] * ew[e] * dinv[d];
  const float2* hv = (const float2*)(h + (size_t)s * HDIM);
  float2 v = hv[lane];
  float* ap = agg + (size_t)d * HDIM + lane * 2;
  unsafeAtomicAdd(ap,     norm * v.x);
  unsafeAtomicAdd(ap + 1, norm * v.y);
}

// ---------------------------------------------------------------------------
// h = elu(agg + dinv^2 * tmp + bias)   (self-loop term fused here)
// ---------------------------------------------------------------------------
__global__ void finalize_elu(const float* __restrict__ agg, const float* __restrict__ tmp,
                             const float* __restrict__ dinv, const float* __restrict__ bias,
                             float* __restrict__ h, int N) {
  int i = blockIdx.x * blockDim.x + threadIdx.x;
  if (i >= N * HDIM) return;
  int n = i >> 6;      // /64
  int f = i & 63;
  float di = dinv[n];
  float v = agg[i] + di * di * tmp[i] + bias[f];
  h[i] = (v > 0.0f) ? v : (expf(v) - 1.0f);
}

// ---------------------------------------------------------------------------
// In-place: out[n,:] = log_softmax(out[n,:] + bias)
// ---------------------------------------------------------------------------
__global__ void logsoftmax_kernel(float* __restrict__ out, const float* __restrict__ bias, int N) {
  int n = blockIdx.x * blockDim.x + threadIdx.x;
  if (n >= N) return;
  float* row = out + (size_t)n * CDIM;
  float mx = -INFINITY;
#pragma unroll
  for (int c = 0; c < CDIM; ++c) mx = fmaxf(mx, row[c] + bias[c]);
  float s = 0.0f;
#pragma unroll
  for (int c = 0; c < CDIM; ++c) s += expf(row[c] + bias[c] - mx);
  float lse = mx + logf(s);
#pragma unroll
  for (int c = 0; c < CDIM; ++c) row[c] = row[c] + bias[c] - lse;
}

// ---------------------------------------------------------------------------
extern "C" void kernel_launch(void* const* d_in, const int* in_sizes, int n_in,
                              void* d_out, int out_size, void* d_ws, size_t ws_size,
                              hipStream_t stream) {
  const float* x   = (const float*)d_in[0];
  const int*   ei  = (const int*)d_in[1];
  const float* ew  = (const float*)d_in[2];
  const float* W0  = (const float*)d_in[3];
  const float* b0  = (const float*)d_in[4];
  const float* W1  = (const float*)d_in[5];
  const float* b1  = (const float*)d_in[6];
  const float* Wl  = (const float*)d_in[7];
  const float* bl  = (const float*)d_in[8];
  float* out = (float*)d_out;

  const int N = in_sizes[0] / F_IN;
  const int E = in_sizes[2];
  const int* src = ei;
  const int* dst = ei + E;

  // Workspace layout (floats): dinv[N] | tmp[N*64] | agg[N*64] | h[N*64]
  float* dinv = (float*)d_ws;
  float* tmp  = dinv + N;
  float* agg  = tmp + (size_t)N * HDIM;
  float* h    = agg + (size_t)N * HDIM;

  const int T = 256;
  const int rowsPerBlock = 16 * (T / 32);       // 8 waves * 16 rows
  const int gemmBlocks   = (N + rowsPerBlock - 1) / rowsPerBlock;
  const int edgeThreads  = E * 32;

  // ---- degree + dinv (deg staged in tmp[0:N], later overwritten by GEMM) ----
  hipMemsetAsync(tmp, 0, (size_t)N * sizeof(float), stream);
  deg_kernel<<<(E + T - 1) / T, T, 0, stream>>>(dst, ew, tmp, E);
  dinv_kernel<<<(N + T - 1) / T, T, 0, stream>>>(tmp, dinv, N);

  // ---- layer 0: tmp = X @ W0 ; agg = scatter ; h = elu(agg + dinv^2*tmp + b0)
  gemm_wmma_f32<4, F_IN><<<gemmBlocks, T, 0, stream>>>(x, W0, tmp, N, HDIM);
  hipMemsetAsync(agg, 0, (size_t)N * HDIM * sizeof(float), stream);
  scatter_kernel<<<(edgeThreads + T - 1) / T, T, 0, stream>>>(src, dst, ew, dinv, tmp, agg, E);
  finalize_elu<<<((N * HDIM) + T - 1) / T, T, 0, stream>>>(agg, tmp, dinv, b0, h, N);

  // ---- layer 1: tmp = h @ W1 ; scatter ; h = elu(...) ----
  gemm_wmma_f32<4, HDIM><<<gemmBlocks, T, 0, stream>>>(h, W1, tmp, N, HDIM);
  hipMemsetAsync(agg, 0, (size_t)N * HDIM * sizeof(float), stream);
  scatter_kernel<<<(edgeThreads + T - 1) / T, T, 0, stream>>>(src, dst, ew, dinv, tmp, agg, E);
  finalize_elu<<<((N * HDIM) + T - 1) / T, T, 0, stream>>>(agg, tmp, dinv, b1, h, N);

  // ---- head: out = h @ Wl ; out = log_softmax(out + bl) ----
  gemm_wmma_f32<3, HDIM><<<gemmBlocks, T, 0, stream>>>(h, Wl, out, N, CDIM);
  logsoftmax_kernel<<<(N + T - 1) / T, T, 0, stream>>>(out, bl, N);
}